// Signature_23639499997478
// MI455X (gfx1250) — compile-verified
//
#include <hip/hip_runtime.h>
#include <hip/hip_bf16.h>

// Path signature, depth 4, path (64, 4096, 6) f32 -> out (64, 1554) f32.
//
// Chen-associative chunking:
//  Kernel 1: one block per (sample, chunk): scan levels 1-3 serially (s3 in
//            registers, s1/s2 parity-double-buffered in LDS, 2 barriers/step),
//            stage 32-step K-panels in LDS (K-pair interleaved so WMMA
//            fragments load as single ds_load_b64), and accumulate the whole
//            level-4 block as three GEMMs with K=chunk via
//            V_WMMA_F32_16X16X4_F32 (exact f32 semantics).
//  Kernel 2: one block per sample folds the 16 chunk signatures with Chen's
//            relation and writes the output.

typedef float v2f __attribute__((ext_vector_type(2)));
typedef float v8f __attribute__((ext_vector_type(8)));

#define NSAMP 64
#define LPATH 4096
#define CDIM  6
#define TSTEPS 4095            // LPATH-1 increments
#define NC    16               // chunks per sample
#define CHUNK 256              // steps per chunk (last chunk has 255)
#define SUB   32               // K-panel depth staged in LDS
#define NSUB  (CHUNK / SUB)    // 8 panels per chunk
#define NTHREADS 288           // 9 wave32 waves
#define NWAVES 9
#define SIGLEN 1554            // 6 + 36 + 216 + 1296

// K-pair interleaved panel addressing: element (k, c) of a panel with S
// columns lives at (k>>1)*(2*S) + 2*c + (k&1).  A lane's WMMA fragment
// (K, K+1) at one column is then an aligned 8-byte pair -> ds_load_b64.
__device__ __forceinline__ int pidx(int k, int c, int S) {
    return ((k >> 1) * (S << 1)) + (c << 1) + (k & 1);
}

// ---- LDS layout (floats) ----
#define OFF_DX   0             // 32 x 16   (B of GEMM1)       512
#define OFF_S1   512           // 32 x 16   (A of GEMM3)       512
#define OFF_E2   1024          // 32 x 48   (B of GEMM2)       1536
#define OFF_S2   2560          // 32 x 48   (A of GEMM2)       1536
#define OFF_E3   4096          // 32 x 224  (B of GEMM3)       7168
#define OFF_P3   11264         // 32 x 224  (A of GEMM1)       7168
#define OFF_S2ST 18432         // s2 state, double buffered: 2 x 40
#define OFF_S1ST 18512         // s1 state, double buffered: 2 x 8
#define OFF_DXC  18528         // dx for whole chunk: 256 x 6 = 1536
#define LDS_FLOATS 20064       // 80,256 bytes (<< 320 KB WGP LDS)

// C-tile staging (reuses panel space after the last WMMA of the chunk):
#define OFF_C3   0             // 16 x 224 = 3584 (over DX/S1/E2/S2 region)
#define OFF_C2   OFF_E3        // 48 x 48  = 2304
#define OFF_C1   OFF_P3        // 224 x 16 = 3584

struct TP { int Ab, As, Bb, Bs, mb, nb, Cb, Cs; };

// 37 WMMA output tiles: 14 for C1 (216x6 pad 224x16), 9 for C2 (36x36 pad
// 48x48), 14 for C3 (6x216 pad 16x224).
__device__ __forceinline__ TP tile_params(int tt) {
    TP p;
    if (tt < 14) {                       // GEMM1: C1 = P3^T * DX
        p.Ab = OFF_P3; p.As = 224; p.Bb = OFF_DX; p.Bs = 16;
        p.mb = tt * 16; p.nb = 0;
        p.Cb = OFF_C1; p.Cs = 16;
    } else if (tt < 23) {                // GEMM2: C2 = S2^T * E2
        int g = tt - 14;
        p.Ab = OFF_S2; p.As = 48; p.Bb = OFF_E2; p.Bs = 48;
        p.mb = (g / 3) * 16; p.nb = (g % 3) * 16;
        p.Cb = OFF_C2; p.Cs = 48;
    } else {                             // GEMM3: C3 = S1^T * E3
        int g = tt - 23;
        p.Ab = OFF_S1; p.As = 16; p.Bb = OFF_E3; p.Bs = 224;
        p.mb = 0; p.nb = g * 16;
        p.Cb = OFF_C3; p.Cs = 224;
    }
    return p;
}

__global__ void sig_chunk_kernel(const float* __restrict__ path,
                                 float* __restrict__ ws) {
    extern __shared__ float sm[];
    const int tid  = threadIdx.x;
    const int lane = tid & 31;
    const int wave = tid >> 5;
    const int half = lane >> 4;       // 0: lanes 0-15, 1: lanes 16-31
    const int lr   = lane & 15;

    const int n  = blockIdx.x / NC;   // sample
    const int ch = blockIdx.x % NC;   // chunk
    const int t0 = ch * CHUNK;
    const int tend = (t0 + CHUNK < TSTEPS) ? (t0 + CHUNK) : TSTEPS;
    const int steps = tend - t0;      // 256 (255 for last chunk)

    // ---- init: zero panels + state; fill dx for the chunk ----
    for (int i = tid; i < OFF_DXC; i += NTHREADS) sm[i] = 0.0f;
    {
        const float* pb = path + ((size_t)n * LPATH + t0) * CDIM;
        for (int i = tid; i < steps * CDIM; i += NTHREADS)
            sm[OFF_DXC + i] = pb[i + CDIM] - pb[i];
    }
    // persistent WMMA accumulators: tile tt = wave + 9*s
    v8f acc[5];
    #pragma unroll
    for (int s = 0; s < 5; ++s) acc[s] = (v8f)0.0f;

    // level-3 state lives in a register of its owning thread (tid < 216)
    float s3_reg = 0.0f;
    const int i3a = tid / 36, i3b = (tid / 6) % 6, i3c = tid % 6; // digits
    __syncthreads();

    for (int sub = 0; sub < NSUB; ++sub) {
        // ---------- scan phase: 32 steps, levels 1..3, 2 barriers/step ------
        for (int j = 0; j < SUB; ++j) {
            const int tl = sub * SUB + j;          // chunk-local step
            const int cur = tl & 1, nxt = cur ^ 1; // state parity buffers
            const bool active = (tl < steps);
            const float* dx = sm + OFF_DXC + tl * CDIM;
            const float* s1c = sm + OFF_S1ST + cur * 8;
            const float* s2c = sm + OFF_S2ST + cur * 40;
            float e3_i = 0.0f;

            // Phase A: store exclusive-prefix rows + e-values
            if (tid < 216) {
                if (active) {
                    e3_i = dx[i3a] * dx[i3b] * dx[i3c] * (1.0f / 6.0f);
                    sm[OFF_E3 + pidx(j, tid, 224)] = e3_i;
                    sm[OFF_P3 + pidx(j, tid, 224)] = s3_reg + 0.25f * e3_i;
                } else {
                    sm[OFF_E3 + pidx(j, tid, 224)] = 0.0f;
                    sm[OFF_P3 + pidx(j, tid, 224)] = 0.0f;
                }
            }
            if (tid < 36) {
                const int a = tid / 6, b = tid % 6;
                const float e2v = active ? dx[a] * dx[b] * 0.5f : 0.0f;
                sm[OFF_E2 + pidx(j, tid, 48)] = e2v;
                sm[OFF_S2 + pidx(j, tid, 48)] = active ? s2c[tid] : 0.0f;
            }
            if (tid < 6) {
                sm[OFF_S1 + pidx(j, tid, 16)] = active ? s1c[tid] : 0.0f;
                sm[OFF_DX + pidx(j, tid, 16)] = active ? dx[tid] : 0.0f;
            }
            __syncthreads();

            // Phase B: advance state (writes go to the other parity buffer)
            if (tid < 216) {
                if (active) {
                    const int ab = tid / 6, bc = tid % 36;
                    s3_reg += s2c[ab] * dx[i3c]
                            + s1c[i3a] * sm[OFF_E2 + pidx(j, bc, 48)]
                            + e3_i;
                }
            } else if (tid < 252) {
                const int i2 = tid - 216;
                const int a = i2 / 6, b = i2 % 6;
                float v = s2c[i2];
                if (active) v += s1c[a] * dx[b] + sm[OFF_E2 + pidx(j, i2, 48)];
                sm[OFF_S2ST + nxt * 40 + i2] = v;
            } else if (tid < 258) {
                const int a = tid - 252;
                sm[OFF_S1ST + nxt * 8 + a] = s1c[a] + (active ? dx[a] : 0.0f);
            }
            __syncthreads();
        }

        // ---------- WMMA phase: 37 tiles over this K=32 panel --------------
        // Wave-uniform control only: EXEC is all-ones around every WMMA.
        // Fragments are single ds_load_b64; next-kt loads are software-
        // pipelined across the current WMMA.
        #pragma unroll
        for (int s = 0; s < 5; ++s) {
            const int tt = wave + NWAVES * s;
            if (tt < 37) {
                const TP p = tile_params(tt);
                const int astep = p.As << 2;  // pair-row stride * 2 kt
                const int bstep = p.Bs << 2;
                int aoff = p.Ab + half * (p.As << 1) + ((p.mb + lr) << 1);
                int boff = p.Bb + half * (p.Bs << 1) + ((p.nb + lr) << 1);
                v2f a = *(const v2f*)(sm + aoff);
                v2f b = *(const v2f*)(sm + boff);
                #pragma unroll
                for (int kt = 0; kt < SUB / 4; ++kt) {
                    v2f an = a, bn = b;
                    if (kt < SUB / 4 - 1) {
                        an = *(const v2f*)(sm + aoff + astep);
                        bn = *(const v2f*)(sm + boff + bstep);
                        aoff += astep; boff += bstep;
                    }
                    acc[s] = __builtin_amdgcn_wmma_f32_16x16x4_f32(
                        false, a, false, b, (short)0, acc[s], false, false);
                    a = an; b = bn;
                }
            }
        }
        __syncthreads();   // panel buffers are rewritten next iteration
    }

    // ---------- stage C tiles to LDS (deterministic, no atomics) ----------
    #pragma unroll
    for (int s = 0; s < 5; ++s) {
        const int tt = wave + NWAVES * s;
        if (tt < 37) {
            const TP p = tile_params(tt);
            #pragma unroll
            for (int v = 0; v < 8; ++v) {
                const int M = p.mb + v + 8 * half;
                const int N = p.nb + lr;
                sm[p.Cb + M * p.Cs + N] = acc[s][v];
            }
        }
    }
    __syncthreads();

    // ---------- write chunk signature: ws[(n*NC + ch) * 1554 + ...] --------
    // Full CHUNK of parity steps always executes -> final state parity is 0.
    float* out = ws + ((size_t)n * NC + ch) * SIGLEN;
    if (tid < 6)   out[tid]      = sm[OFF_S1ST + tid];
    if (tid < 36)  out[6 + tid]  = sm[OFF_S2ST + tid];
    if (tid < 216) out[42 + tid] = s3_reg;
    for (int i = tid; i < 1296; i += NTHREADS) {
        const float c1 = sm[OFF_C1 + (i / 6)   * 16  + (i % 6)];
        const float c2 = sm[OFF_C2 + (i / 36)  * 48  + (i % 36)];
        const float c3 = sm[OFF_C3 + (i / 216) * 224 + (i % 216)];
        out[258 + i] = c1 + c2 + c3;
    }
}

// ---- Kernel 2: fold NC chunk signatures per sample with Chen's relation ----
__global__ void chen_combine_kernel(const float* __restrict__ ws,
                                    float* __restrict__ out) {
    __shared__ float bufA[SIGLEN];
    __shared__ float bufB[SIGLEN];
    __shared__ float ebuf[SIGLEN];
    const int tid = threadIdx.x;
    const int n = blockIdx.x;

    float* cur = bufA;
    float* nxt = bufB;
    for (int i = tid; i < SIGLEN; i += blockDim.x) cur[i] = 0.0f;
    __syncthreads();

    for (int c = 0; c < NC; ++c) {
        const float* e = ws + ((size_t)n * NC + c) * SIGLEN;
        for (int i = tid; i < SIGLEN; i += blockDim.x) ebuf[i] = e[i];
        __syncthreads();
        for (int i = tid; i < SIGLEN; i += blockDim.x) {
            float v;
            if (i < 6) {
                v = cur[i] + ebuf[i];
            } else if (i < 42) {
                const int k = i - 6, a = k / 6, b = k % 6;
                v = cur[i] + ebuf[i] + cur[a] * ebuf[b];
            } else if (i < 258) {
                const int k = i - 42;
                const int a = k / 36, bc = k % 36, ab = k / 6, cc = k % 6;
                v = cur[i] + ebuf[i]
                  + cur[a] * ebuf[6 + bc]
                  + cur[6 + ab] * ebuf[cc];
            } else {
                const int k = i - 258;
                const int a = k / 216, bcd = k % 216;
                const int ab = k / 36, cd = k % 36;
                const int abc = k / 6, d = k % 6;
                v = cur[i] + ebuf[i]
                  + cur[a] * ebuf[42 + bcd]
                  + cur[6 + ab] * ebuf[6 + cd]
                  + cur[42 + abc] * ebuf[d];
            }
            nxt[i] = v;
        }
        __syncthreads();
        float* t = cur; cur = nxt; nxt = t;
    }
    for (int i = tid; i < SIGLEN; i += blockDim.x)
        out[(size_t)n * SIGLEN + i] = cur[i];
}

extern "C" void kernel_launch(void* const* d_in, const int* in_sizes, int n_in,
                              void* d_out, int out_size, void* d_ws, size_t ws_size,
                              hipStream_t stream) {
    const float* path = (const float*)d_in[0];   // (64, 4096, 6) f32
    float* ws  = (float*)d_ws;                   // needs 64*16*1554*4 = 6.4 MB
    float* out = (float*)d_out;                  // (64, 1554) f32

    const size_t shmem = (size_t)LDS_FLOATS * sizeof(float);  // ~80 KB dynamic LDS
    sig_chunk_kernel<<<dim3(NSAMP * NC), dim3(NTHREADS), shmem, stream>>>(path, ws);
    chen_combine_kernel<<<dim3(NSAMP), dim3(256), 0, stream>>>(ws, out);
}